// MyModel_61933428416344
// MI455X (gfx1250) — compile-verified
//
#include <hip/hip_runtime.h>

// is_sorted check: out = 1.0f iff every row of x (64 rows x 1048576 cols,
// row-major) is non-decreasing. Pure streaming reduction: 256 MB read once,
// ~11 us floor at 23.3 TB/s. No matrix FLOPs -> no WMMA; optimized around
// wave32 B128 non-temporal loads (4 in flight per thread), cross-lane
// shuffle for the chunk-boundary pair, and an agent-scope early-exit flag.

typedef __attribute__((ext_vector_type(4))) float v4f;

#define ROWLEN_FLOATS    (1u << 20)              // 1048576 floats per row
#define CHUNKS_PER_ROW   (ROWLEN_FLOATS / 4u)    // 262144 float4 chunks per row
#define TOTAL_CHUNKS     (1u << 24)              // 64 rows * 262144
#define BLOCK_THREADS    256u
#define VEC              4u                      // float4 chunks per thread
#define CHUNKS_PER_BLOCK (VEC * BLOCK_THREADS)   // 1024
#define GRID_BLOCKS      (TOTAL_CHUNKS / CHUNKS_PER_BLOCK)  // 16384, exact fit

__global__ void issorted_init_kernel(float* __restrict__ out,
                                     int* __restrict__ flag) {
    // Reset result + early-exit flag every call (harness does not re-poison).
    *out = 1.0f;
    __hip_atomic_store(flag, 1, __ATOMIC_RELAXED, __HIP_MEMORY_SCOPE_AGENT);
}

__global__ __launch_bounds__(BLOCK_THREADS)
void issorted_scan_kernel(const float* __restrict__ x,
                          float* __restrict__ out,
                          int* __restrict__ flag) {
    __shared__ int s_state;  // phase 1: skip?  phase 2: block-found-violation?

    // Early exit: if another block already proved "not sorted", skip all loads.
    if (threadIdx.x == 0) {
        int f = __hip_atomic_load(flag, __ATOMIC_RELAXED, __HIP_MEMORY_SCOPE_AGENT);
        s_state = (f == 0) ? 1 : 0;
    }
    __syncthreads();
    if (s_state != 0) return;          // uniform across block
    // s_state is 0 here; reuse it as the block "violation found" flag.

    const unsigned base = blockIdx.x * CHUNKS_PER_BLOCK + threadIdx.x;
    const unsigned lane = threadIdx.x & 31u;

    // Issue VEC independent 16B NT loads back-to-back: 4 cachelines in flight
    // per wave before the first dependent use (partial loadcnt waits drain
    // them while compare work proceeds).
    unsigned c[VEC];
    v4f      v[VEC];
#pragma unroll
    for (unsigned k = 0; k < VEC; ++k) {
        c[k] = base + k * BLOCK_THREADS;                  // coalesced per wave
        v[k] = __builtin_nontemporal_load((const v4f*)x + c[k]);
    }

    bool bad = false;
#pragma unroll
    for (unsigned k = 0; k < VEC; ++k) {
        // Cross-chunk pair (4c+3, 4c+4) is invalid exactly at a row boundary;
        // the very last chunk of the array is also a row boundary, so no OOB.
        const bool crossValid = (((c[k] + 1u) & (CHUNKS_PER_ROW - 1u)) != 0u);

        // Neighbor's first element via wave32 shuffle (LDS-permute hw, no LDS
        // memory traffic); lane 31 fetches the one element shuffle can't give.
        float nx = __shfl_down(v[k].x, 1, 32);
        if (lane == 31u && crossValid) {
            nx = __builtin_nontemporal_load(x + ((size_t)c[k] * 4u + 4u));
        }

        bad |= (v[k].x > v[k].y) | (v[k].y > v[k].z) | (v[k].z > v[k].w) |
               (crossValid & (v[k].w > nx));
    }

    if (bad) s_state = 1;              // racing same-value LDS stores: fine
    __syncthreads();

    if (threadIdx.x == 0 && s_state != 0) {
        __hip_atomic_store(flag, 0, __ATOMIC_RELAXED, __HIP_MEMORY_SCOPE_AGENT);
        *out = 0.0f;                   // all writers write the same value
    }
}

extern "C" void kernel_launch(void* const* d_in, const int* in_sizes, int n_in,
                              void* d_out, int out_size, void* d_ws, size_t ws_size,
                              hipStream_t stream) {
    const float* x   = (const float*)d_in[0];
    float*       out = (float*)d_out;
    int*         flg = (int*)d_ws;

    issorted_init_kernel<<<1, 1, 0, stream>>>(out, flg);
    issorted_scan_kernel<<<GRID_BLOCKS, BLOCK_THREADS, 0, stream>>>(x, out, flg);
}